// ChamferDistance_72026601554617
// MI455X (gfx1250) — compile-verified
//
#include <hip/hip_runtime.h>
#include <math.h>

typedef float v2f __attribute__((ext_vector_type(2)));
typedef float v4f __attribute__((ext_vector_type(4)));
typedef float v8f __attribute__((ext_vector_type(8)));

#define NPTS   8192
#define NBATCH 8
#define MTILES_PER_BLOCK 8                        // 8 waves/block, one 16-row tile per wave
#define WAVES_PER_PASS (NBATCH * (NPTS / 16))     // 4096
#define STAGE_FLOATS ((size_t)NBATCH * NPTS * 4)  // per staged array

// raw v_min_num_f32 without fminf's sNaN canonicalize (inputs are never NaN)
__device__ __forceinline__ float fmin_raw(float a, float b) {
    float r;
    asm("v_min_num_f32 %0, %1, %2" : "=v"(r) : "v"(a), "v"(b));
    return r;
}

// Stage both arrays as float4 (-2x, -2y, -2z, ||p||^2): the WMMA B-fragment then
// is one b64 load per lane (lanes 0-15 take .xy = K0,K1; lanes 16-31 take .zw = K2,K3).
__global__ __launch_bounds__(256) void chamfer_stage_kernel(
    const float* __restrict__ pred, const float* __restrict__ gt,
    float* __restrict__ staged, int npts)   // staged: [2*npts, 4]
{
    int i = blockIdx.x * blockDim.x + threadIdx.x;
    if (i >= 2 * npts) return;
    const float* src = (i < npts) ? (pred + (size_t)i * 3)
                                  : (gt + (size_t)(i - npts) * 3);
    const float x = src[0], y = src[1], z = src[2];
    v4f o;
    o.x = -2.0f * x;
    o.y = -2.0f * y;
    o.z = -2.0f * z;
    o.w = x * x + y * y + z * z;
    *(v4f*)(staged + (size_t)i * 4) = o;
}

// One wave owns a 16-row tile of "A" points and scans all 16-col tiles of "B"
// points, keeping running row-mins of D[m,n] = ||b_n||^2 - 2 a_m . b_n via
// V_WMMA_F32_16X16X4_F32 (K padded 3->4; affine terms folded into K=3).
// A fragment: lanes 0-15 (K0,K1)=(x,y); lanes 16-31 (K2,K3)=(z,1).
// B fragment pre-staged: lanes 0-15 (-2x,-2y); lanes 16-31 (-2z,||g||^2).
// blockIdx.z selects the direction (0: pred->gt, 1: gt->pred); both directions
// run in one launch so 8192 waves are resident simultaneously.
__global__ __launch_bounds__(256) void chamfer_pass_staged_kernel(
    const float* __restrict__ pred, const float* __restrict__ gt,
    const float* __restrict__ staged,   // [2*B*N, 4]: pred staged, then gt staged
    float* __restrict__ ws)
{
    const int dir   = blockIdx.z;                 // 0: rows=pred, cols=gt ; 1: rows=gt, cols=pred
    const int lane  = threadIdx.x & 31;
    const int wave  = threadIdx.x >> 5;
    const int batch = blockIdx.y;
    const int m0    = blockIdx.x * (16 * MTILES_PER_BLOCK) + wave * 16;
    const int j     = lane & 15;
    const bool hi   = lane >= 16;

    const float* Apts    = dir ? gt : pred;
    const float* Bstaged = staged + (dir ? 0 : STAGE_FLOATS);  // cols: dir0 -> staged gt

    const float* ap = Apts + ((size_t)batch * NPTS + (size_t)(m0 + j)) * 3;
    const float ax = ap[0], ay = ap[1], az = ap[2];
    const float p2 = ax * ax + ay * ay + az * az;

    v2f afrag;
    afrag.x = hi ? az   : ax;
    afrag.y = hi ? 1.0f : ay;

    float rmin[8];
#pragma unroll
    for (int r = 0; r < 8; ++r) rmin[r] = 3.0e38f;

    // per-lane staged pointer: point (n0 + j), floats [0,1] or [2,3]
    const float* sb = Bstaged + (size_t)batch * NPTS * 4 + (size_t)j * 4 + (hi ? 2 : 0);

#pragma unroll 4
    for (int n0 = 0; n0 < NPTS; n0 += 16) {
        v2f bfrag = *(const v2f*)(sb + (size_t)n0 * 4);
        v8f cz = {0.f, 0.f, 0.f, 0.f, 0.f, 0.f, 0.f, 0.f};
        v8f d = __builtin_amdgcn_wmma_f32_16x16x4_f32(
            false, afrag, false, bfrag, (short)0, cz, false, false);
#pragma unroll
        for (int r = 0; r < 8; ++r) rmin[r] = fmin_raw(rmin[r], d[r]);
    }

    // butterfly min across the 16 lanes of each half (masks < 16 stay in-half)
#pragma unroll
    for (int mask = 1; mask <= 8; mask <<= 1) {
#pragma unroll
        for (int r = 0; r < 8; ++r)
            rmin[r] = fmin_raw(rmin[r], __shfl_xor(rmin[r], mask, 32));
    }

    float s = rmin[0] + rmin[1] + rmin[2] + rmin[3]
            + rmin[4] + rmin[5] + rmin[6] + rmin[7];

    // lanes 0/16 inject their half's min-sum; lanes 0..15 inject ||p||^2 of rows m0..m0+15
    float c = (j == 0 ? s : 0.0f) + (!hi ? p2 : 0.0f);
#pragma unroll
    for (int mask = 16; mask >= 1; mask >>= 1)
        c += __shfl_xor(c, mask, 32);

    if (lane == 0) {
        const int waveId = (batch * (int)gridDim.x + (int)blockIdx.x) * MTILES_PER_BLOCK + wave;
        ws[dir * WAVES_PER_PASS + waveId] = c * (1.0f / ((float)NBATCH * (float)NPTS));
    }
}

// Fallback (small ws): build the B fragment in-loop from raw points; both
// directions still merged into one launch via blockIdx.z.
__global__ __launch_bounds__(256) void chamfer_pass_kernel(
    const float* __restrict__ pred, const float* __restrict__ gt,
    float* __restrict__ ws)
{
    const int dir   = blockIdx.z;
    const int lane  = threadIdx.x & 31;
    const int wave  = threadIdx.x >> 5;
    const int batch = blockIdx.y;
    const int m0    = blockIdx.x * (16 * MTILES_PER_BLOCK) + wave * 16;
    const int j     = lane & 15;
    const bool hi   = lane >= 16;

    const float* Apts = dir ? gt : pred;
    const float* Bpts = dir ? pred : gt;

    const float* ap = Apts + ((size_t)batch * NPTS + (size_t)(m0 + j)) * 3;
    const float ax = ap[0], ay = ap[1], az = ap[2];
    const float p2 = ax * ax + ay * ay + az * az;

    v2f afrag;
    afrag.x = hi ? az   : ax;
    afrag.y = hi ? 1.0f : ay;

    float rmin[8];
#pragma unroll
    for (int r = 0; r < 8; ++r) rmin[r] = 3.0e38f;

    const float* gbase = Bpts + (size_t)batch * NPTS * 3;

#pragma unroll 2
    for (int n0 = 0; n0 < NPTS; n0 += 16) {
        const float* gp = gbase + (size_t)(n0 + j) * 3;
        const float gx = gp[0], gy = gp[1], gz = gp[2];
        const float g2 = gx * gx + gy * gy + gz * gz;
        v2f bfrag;
        bfrag.x = hi ? (-2.0f * gz) : (-2.0f * gx);
        bfrag.y = hi ? g2           : (-2.0f * gy);
        v8f cz = {0.f, 0.f, 0.f, 0.f, 0.f, 0.f, 0.f, 0.f};
        v8f d = __builtin_amdgcn_wmma_f32_16x16x4_f32(
            false, afrag, false, bfrag, (short)0, cz, false, false);
#pragma unroll
        for (int r = 0; r < 8; ++r) rmin[r] = fmin_raw(rmin[r], d[r]);
    }

#pragma unroll
    for (int mask = 1; mask <= 8; mask <<= 1) {
#pragma unroll
        for (int r = 0; r < 8; ++r)
            rmin[r] = fmin_raw(rmin[r], __shfl_xor(rmin[r], mask, 32));
    }

    float s = rmin[0] + rmin[1] + rmin[2] + rmin[3]
            + rmin[4] + rmin[5] + rmin[6] + rmin[7];
    float c = (j == 0 ? s : 0.0f) + (!hi ? p2 : 0.0f);
#pragma unroll
    for (int mask = 16; mask >= 1; mask >>= 1)
        c += __shfl_xor(c, mask, 32);

    if (lane == 0) {
        const int waveId = (batch * (int)gridDim.x + (int)blockIdx.x) * MTILES_PER_BLOCK + wave;
        ws[dir * WAVES_PER_PASS + waveId] = c * (1.0f / ((float)NBATCH * (float)NPTS));
    }
}

// Deterministic single-block reduction of all wave partials -> scalar output.
__global__ __launch_bounds__(256) void chamfer_reduce_kernel(
    const float* __restrict__ ws, float* __restrict__ out, int n)
{
    __shared__ float sm[256];
    float s = 0.0f;
    for (int i = threadIdx.x; i < n; i += 256) s += ws[i];
    sm[threadIdx.x] = s;
    __syncthreads();
    for (int k = 128; k >= 1; k >>= 1) {
        if ((int)threadIdx.x < k) sm[threadIdx.x] += sm[threadIdx.x + k];
        __syncthreads();
    }
    if (threadIdx.x == 0) out[0] = sm[0];
}

extern "C" void kernel_launch(void* const* d_in, const int* in_sizes, int n_in,
                              void* d_out, int out_size, void* d_ws, size_t ws_size,
                              hipStream_t stream) {
    const float* pred = (const float*)d_in[0];
    const float* gt   = (const float*)d_in[1];
    float* out = (float*)d_out;
    float* ws  = (float*)d_ws;

    dim3 grid(NPTS / (16 * MTILES_PER_BLOCK), NBATCH, 2);  // (64, 8, 2): both directions
    dim3 block(256);

    const size_t needStaged = (2 * STAGE_FLOATS + 2 * WAVES_PER_PASS) * sizeof(float);

    if (ws_size >= needStaged) {
        float* staged   = ws;                      // [2*B*N, 4]: pred staged, then gt staged
        float* partials = ws + 2 * STAGE_FLOATS;   // [2 * 4096]
        const int npts = NBATCH * NPTS;

        chamfer_stage_kernel<<<dim3((2 * npts + 255) / 256), block, 0, stream>>>(
            pred, gt, staged, npts);
        chamfer_pass_staged_kernel<<<grid, block, 0, stream>>>(pred, gt, staged, partials);
        chamfer_reduce_kernel<<<dim3(1), block, 0, stream>>>(partials, out, 2 * WAVES_PER_PASS);
    } else {
        float* partials = ws;  // [2 * 4096] = 32 KB
        chamfer_pass_kernel<<<grid, block, 0, stream>>>(pred, gt, partials);
        chamfer_reduce_kernel<<<dim3(1), block, 0, stream>>>(partials, out, 2 * WAVES_PER_PASS);
    }
}